// SMPL_Layer_81947976008375
// MI455X (gfx1250) — compile-verified
//
#include <hip/hip_runtime.h>
#include <math.h>

// ---------------------------------------------------------------------------
// SMPL forward for MI455X (gfx1250, wave32, WMMA + TDM).
// B=1024, V=6890, J=24.  fp32 end-to-end using V_WMMA_F32_16X16X4_F32,
// A-tile staged to LDS by the Tensor Data Mover (tensor_load_to_lds).
// ---------------------------------------------------------------------------

#define NUM_JOINTS 24
#define NUM_VERTS  6890
#define NROWS      20670        // NUM_VERTS*3 rows of the blend matrices
#define BATCH      1024
#define KPOSE      207          // (J-1)*9
#define KTOT       220          // 207 pose + 10 betas + 1 template + 2 pad
#define VT_TILE    16           // vertices per block  (48 output rows)
#define BT_TILE    64           // batches per block   (4 waves x 16 cols)

typedef float        v2f __attribute__((ext_vector_type(2)));
typedef float        v8f __attribute__((ext_vector_type(8)));
typedef unsigned int v4u __attribute__((ext_vector_type(4)));
typedef int          v4i __attribute__((ext_vector_type(4)));
typedef int          v8i __attribute__((ext_vector_type(8)));

__device__ __constant__ int c_parents[NUM_JOINTS] =
    {-1, 0, 0, 0, 1, 2, 3, 4, 5, 6, 7, 8, 9, 9, 9, 12, 13, 14, 16, 17, 18, 19, 20, 21};

// ---------------------------------------------------------------------------
// Kernel 1: batch-independent joint-regressor factoring.
//   JS[(j*3+c)*10+s] = sum_v Jreg[j,v] * shapedirs[v,c,s]
//   JV[j*3+c]        = sum_v Jreg[j,v] * v_template[v,c]
// ---------------------------------------------------------------------------
__global__ __launch_bounds__(256)
void joint_reg_precompute(const float* __restrict__ Jreg,
                          const float* __restrict__ shapedirs,
                          const float* __restrict__ v_template,
                          float* __restrict__ JS, float* __restrict__ JV) {
    const int j   = blockIdx.x;
    const int tid = threadIdx.x;
    float acc[33];
#pragma unroll
    for (int o = 0; o < 33; ++o) acc[o] = 0.f;

    for (int v = tid; v < NUM_VERTS; v += 256) {
        const float jr = Jreg[j * NUM_VERTS + v];
#pragma unroll
        for (int o = 0; o < 30; ++o) acc[o] += jr * shapedirs[v * 30 + o];
        acc[30] += jr * v_template[v * 3 + 0];
        acc[31] += jr * v_template[v * 3 + 1];
        acc[32] += jr * v_template[v * 3 + 2];
    }

    __shared__ float red[256];
    for (int o = 0; o < 33; ++o) {
        red[tid] = acc[o];
        __syncthreads();
        for (int s = 128; s > 0; s >>= 1) {
            if (tid < s) red[tid] += red[tid + s];
            __syncthreads();
        }
        if (tid == 0) {
            if (o < 30) JS[j * 30 + o] = red[0];
            else        JV[j * 3 + (o - 30)] = red[0];
        }
        __syncthreads();
    }
}

// ---------------------------------------------------------------------------
// Kernel 2: one wave32 per batch element.
//   Rodrigues -> B' operand (transposed [k][b]), th_j, kinematic chain,
//   corrected G2 (t_corr folded), th_jtr output.
// ---------------------------------------------------------------------------
__global__ __launch_bounds__(32)
void batch_transforms(const float* __restrict__ pose,
                      const float* __restrict__ betas,
                      const float* __restrict__ trans,
                      const float* __restrict__ JS,
                      const float* __restrict__ JV,
                      float* __restrict__ bvec,   // [KTOT][BATCH]
                      float* __restrict__ g2,     // [BATCH][24][16]
                      float* __restrict__ jtr) {  // [BATCH][24][3]
    const int b    = blockIdx.x;
    const int lane = threadIdx.x;

    __shared__ float R[NUM_JOINTS][9];
    __shared__ float Jt[NUM_JOINTS][3];
    __shared__ float G[NUM_JOINTS][16];

    // --- Rodrigues, one joint per lane ---
    if (lane < NUM_JOINTS) {
        const float ax = pose[b * 72 + lane * 3 + 0];
        const float ay = pose[b * 72 + lane * 3 + 1];
        const float az = pose[b * 72 + lane * 3 + 2];
        const float th = sqrtf(ax * ax + ay * ay + az * az + 1e-8f);
        const float inv = 1.f / th;
        const float kx = ax * inv, ky = ay * inv, kz = az * inv;
        const float c = cosf(th), s = sinf(th), ic = 1.f - c;
        R[lane][0] = c + ic * kx * kx;
        R[lane][1] = ic * kx * ky - s * kz;
        R[lane][2] = ic * kx * kz + s * ky;
        R[lane][3] = ic * ky * kx + s * kz;
        R[lane][4] = c + ic * ky * ky;
        R[lane][5] = ic * ky * kz - s * kx;
        R[lane][6] = ic * kz * kx - s * ky;
        R[lane][7] = ic * kz * ky + s * kx;
        R[lane][8] = c + ic * kz * kz;
    }
    __syncthreads();

    // --- B' operand, transposed for coalesced WMMA-B loads ---
    if (lane >= 1 && lane < NUM_JOINTS) {
#pragma unroll
        for (int e = 0; e < 9; ++e) {
            const float id = (e == 0 || e == 4 || e == 8) ? 1.f : 0.f;
            bvec[((lane - 1) * 9 + e) * BATCH + b] = R[lane][e] - id;
        }
    }
    if (lane < 10) bvec[(KPOSE + lane) * BATCH + b] = betas[b * 10 + lane];
    if (lane == 0) {
        bvec[217 * BATCH + b] = 1.f;
        bvec[218 * BATCH + b] = 0.f;
        bvec[219 * BATCH + b] = 0.f;
    }

    // --- th_j from factored regressor ---
    for (int o = lane; o < 72; o += 32) {
        float acc = JV[o];
#pragma unroll
        for (int s = 0; s < 10; ++s) acc += JS[o * 10 + s] * betas[b * 10 + s];
        Jt[o / 3][o % 3] = acc;
    }
    __syncthreads();

    // --- kinematic chain: 16 lanes, one (m,n) element each ---
    if (lane < 16) {
        const int m = lane >> 2, n = lane & 3;
        G[0][lane] = (m < 3) ? ((n < 3) ? R[0][m * 3 + n] : Jt[0][m])
                             : ((n == 3) ? 1.f : 0.f);
    }
    __syncthreads();
    for (int i = 1; i < NUM_JOINTS; ++i) {
        const int p = c_parents[i];
        if (lane < 16) {
            const int m = lane >> 2, n = lane & 3;
            float acc = 0.f;
#pragma unroll
            for (int q = 0; q < 4; ++q) {
                float relqn;
                if (q < 3) relqn = (n < 3) ? R[i][q * 3 + n] : (Jt[i][q] - Jt[p][q]);
                else       relqn = (n == 3) ? 1.f : 0.f;
                acc += G[p][m * 4 + q] * relqn;
            }
            G[i][lane] = acc;
        }
        __syncthreads();
    }

    // --- G2 = G with translation column corrected by G @ [th_j,0] ---
    float* g2b = g2 + (size_t)b * NUM_JOINTS * 16;
    for (int o = lane; o < NUM_JOINTS * 4; o += 32) {
        const int i = o >> 2, m = o & 3;
        const float tc = G[i][m * 4 + 0] * Jt[i][0] +
                         G[i][m * 4 + 1] * Jt[i][1] +
                         G[i][m * 4 + 2] * Jt[i][2];
        g2b[i * 16 + m * 4 + 0] = G[i][m * 4 + 0];
        g2b[i * 16 + m * 4 + 1] = G[i][m * 4 + 1];
        g2b[i * 16 + m * 4 + 2] = G[i][m * 4 + 2];
        g2b[i * 16 + m * 4 + 3] = G[i][m * 4 + 3] - tc;
    }
    // --- th_jtr = G[:, :3, 3] + trans ---
    for (int o = lane; o < 72; o += 32) {
        const int i = o / 3, c = o % 3;
        jtr[b * 72 + o] = G[i][c * 4 + 3] + trans[b * 3 + c];
    }
}

// ---------------------------------------------------------------------------
// Kernel 3: fused pose/shape blend GEMM (WMMA f32 16x16x4) + LBS skinning.
// A-tile (48 x 220, posedirs part) staged to LDS by the Tensor Data Mover;
// tile_dim0=220 > tensor_dim0=207 zero-fills the padding columns in HW.
// ---------------------------------------------------------------------------
__global__ __launch_bounds__(128)
void fused_pose_skin(const float* __restrict__ posedirs,
                     const float* __restrict__ shapedirs,
                     const float* __restrict__ v_template,
                     const float* __restrict__ weights,
                     const float* __restrict__ trans,
                     const float* __restrict__ bvec,   // [KTOT][BATCH]
                     const float* __restrict__ g2,     // [BATCH][24][16]
                     float* __restrict__ verts) {      // [BATCH][V][3]
    __shared__ float As[48 * KTOT];           // A' tile: 48 rows x K=220
    __shared__ float Ps[48 * (BT_TILE + 1)];  // v_posed tile: 48 x 64 (+pad)

    const int tid  = threadIdx.x;
    const int v0   = blockIdx.x * VT_TILE;
    const int bblk = blockIdx.y * BT_TILE;

    // ---- TDM: DMA posedirs rows [3*v0, 3*v0+48) x 207 into LDS ----------
    if (tid < 32) {
        const unsigned lds_base = (unsigned)(uintptr_t)&As[0];
        const unsigned long long ga =
            (unsigned long long)(uintptr_t)(posedirs + (size_t)(v0 * 3) * KPOSE);
        const unsigned rows_left = (unsigned)(NROWS - v0 * 3);  // OOB rows -> 0

        v4u g0;
        g0.x = 1u;                                   // count=1 (valid), load
        g0.y = lds_base;                             // D#.lds_addr (bytes)
        g0.z = (unsigned)(ga & 0xFFFFFFFFu);         // global_addr[31:0]
        g0.w = (unsigned)((ga >> 32) & 0x01FFFFFFu)  // global_addr[56:32]
               | (2u << 30);                         // type = 2 ("image")

        v8i g1;
        g1[0] = (int)(2u << 16);                     // data_size=4B, no mcast
        g1[1] = (int)((unsigned)KPOSE << 16);        // tensor_dim0[15:0]=207
        g1[2] = (int)(rows_left << 16);              // tensor_dim1[15:0]
        g1[3] = (int)((rows_left >> 16) |            // tensor_dim1[31:16]
                      ((unsigned)KTOT << 16));       // tile_dim0 = 220
        g1[4] = (int)48;                             // tile_dim1=48, tile_dim2=0
        g1[5] = (int)KPOSE;                          // tensor_dim0_stride = 207
        g1[6] = 0;                                   // stride hi / dim1_stride lo
        g1[7] = 0;                                   // tensor_dim1_stride hi

        const v4i z4 = {0, 0, 0, 0};
#if __has_include(<hip/amd_detail/amd_gfx1250_TDM.h>)
        const v8i z8 = {0, 0, 0, 0, 0, 0, 0, 0};
        __builtin_amdgcn_tensor_load_to_lds(g0, g1, z4, z4, z8, 0);
#else
        __builtin_amdgcn_tensor_load_to_lds(g0, g1, z4, z4, 0);
#endif
        __builtin_amdgcn_s_wait_tensorcnt(0);
    }
    // prefetch the weights tile needed by the skinning epilogue
    if (tid < 12)
        __builtin_prefetch((const char*)(weights + (size_t)v0 * NUM_JOINTS) +
                               tid * 128, 0, 0);
    __syncthreads();

    // ---- patch augmented columns: [207..216]=shapedirs, [217]=template ---
    for (int idx = tid; idx < 48 * 13; idx += 128) {
        const int r = idx / 13, kk = idx - r * 13;   // kk in [0,13)
        const int v = v0 + (r / 3);
        const int c = r % 3;
        float val = 0.f;
        if (v < NUM_VERTS) {
            if (kk < 10)       val = shapedirs[(v * 3 + c) * 10 + kk];
            else if (kk == 10) val = v_template[v * 3 + c];
        }
        As[r * KTOT + (KPOSE + kk)] = val;
    }
    __syncthreads();

    // ---- WMMA f32 16x16x4 main loop (wave32 fragment layouts) ----
    const int wave = tid >> 5;          // 0..3
    const int lane = tid & 31;
    const int half = lane >> 4;         // K-half select
    const int l16  = lane & 15;
    const int bw   = bblk + wave * 16;  // this wave's 16-batch column tile

    v8f acc0 = {}, acc1 = {}, acc2 = {};
    for (int kt = 0; kt < KTOT; kt += 4) {
        const int k = kt + half * 2;
        v2f bf;                               // B: 4x16, coalesced [k][b] loads
        bf.x = bvec[k * BATCH + bw + l16];
        bf.y = bvec[(k + 1) * BATCH + bw + l16];
        v2f a0, a1, a2;                       // A: 16x4 per row-block from LDS
        a0.x = As[l16 * KTOT + k];        a0.y = As[l16 * KTOT + k + 1];
        a1.x = As[(16 + l16) * KTOT + k]; a1.y = As[(16 + l16) * KTOT + k + 1];
        a2.x = As[(32 + l16) * KTOT + k]; a2.y = As[(32 + l16) * KTOT + k + 1];
        acc0 = __builtin_amdgcn_wmma_f32_16x16x4_f32(false, a0, false, bf,
                                                     (short)0, acc0, false, false);
        acc1 = __builtin_amdgcn_wmma_f32_16x16x4_f32(false, a1, false, bf,
                                                     (short)0, acc1, false, false);
        acc2 = __builtin_amdgcn_wmma_f32_16x16x4_f32(false, a2, false, bf,
                                                     (short)0, acc2, false, false);
    }

    // ---- spill v_posed tile to LDS (C/D layout: VGPR g -> M = g + half*8) ----
    const int col = wave * 16 + l16;
#pragma unroll
    for (int g = 0; g < 8; ++g) {
        const int row = g + half * 8;
        Ps[(0 * 16 + row) * (BT_TILE + 1) + col] = acc0[g];
        Ps[(1 * 16 + row) * (BT_TILE + 1) + col] = acc1[g];
        Ps[(2 * 16 + row) * (BT_TILE + 1) + col] = acc2[g];
    }
    __syncthreads();

    // ---- linear blend skinning: 16v x 64b pairs, 8 per thread ----
    for (int p = tid; p < VT_TILE * BT_TILE; p += 128) {
        const int lb = p & (BT_TILE - 1);
        const int lv = p >> 6;
        const int v  = v0 + lv;
        if (v >= NUM_VERTS) continue;
        const int b = bblk + lb;

        const float x = Ps[(lv * 3 + 0) * (BT_TILE + 1) + lb];
        const float y = Ps[(lv * 3 + 1) * (BT_TILE + 1) + lb];
        const float z = Ps[(lv * 3 + 2) * (BT_TILE + 1) + lb];

        float o0 = 0.f, o1 = 0.f, o2 = 0.f;
        const float* g2b = g2 + (size_t)b * NUM_JOINTS * 16;
        const float* wv  = weights + (size_t)v * NUM_JOINTS;
#pragma unroll 4
        for (int j = 0; j < NUM_JOINTS; ++j) {
            const float wj = wv[j];
            const float* Gm = g2b + j * 16;
            o0 += wj * (Gm[0] * x + Gm[1] * y + Gm[2]  * z + Gm[3]);
            o1 += wj * (Gm[4] * x + Gm[5] * y + Gm[6]  * z + Gm[7]);
            o2 += wj * (Gm[8] * x + Gm[9] * y + Gm[10] * z + Gm[11]);
        }
        float* out = verts + ((size_t)b * NUM_VERTS + v) * 3;
        out[0] = o0 + trans[b * 3 + 0];
        out[1] = o1 + trans[b * 3 + 1];
        out[2] = o2 + trans[b * 3 + 2];
    }
}

// ---------------------------------------------------------------------------
extern "C" void kernel_launch(void* const* d_in, const int* in_sizes, int n_in,
                              void* d_out, int out_size, void* d_ws, size_t ws_size,
                              hipStream_t stream) {
    (void)in_sizes; (void)n_in; (void)out_size; (void)ws_size;
    const float* pose       = (const float*)d_in[0];
    const float* betas      = (const float*)d_in[1];
    const float* trans      = (const float*)d_in[2];
    const float* v_template = (const float*)d_in[3];
    const float* shapedirs  = (const float*)d_in[4];
    const float* posedirs   = (const float*)d_in[5];
    const float* Jreg       = (const float*)d_in[6];
    const float* weights    = (const float*)d_in[7];

    float* verts = (float*)d_out;
    float* jtr   = verts + (size_t)BATCH * NUM_VERTS * 3;

    float* ws   = (float*)d_ws;
    float* JS   = ws;                               // 720
    float* JV   = ws + 768;                         // 72
    float* bvec = ws + 1024;                        // KTOT*BATCH = 225280
    float* g2   = ws + 1024 + KTOT * BATCH;         // 393216

    joint_reg_precompute<<<NUM_JOINTS, 256, 0, stream>>>(Jreg, shapedirs,
                                                         v_template, JS, JV);
    batch_transforms<<<BATCH, 32, 0, stream>>>(pose, betas, trans, JS, JV,
                                               bvec, g2, jtr);
    fused_pose_skin<<<dim3((NUM_VERTS + VT_TILE - 1) / VT_TILE, BATCH / BT_TILE),
                      128, 0, stream>>>(posedirs, shapedirs, v_template, weights,
                                        trans, bvec, g2, verts);
}